// MMD_loss_25185688223796
// MI455X (gfx1250) — compile-verified
//
#include <hip/hip_runtime.h>

typedef __attribute__((ext_vector_type(2))) float v2f;
typedef __attribute__((ext_vector_type(8))) float v8f;

#define NROWS 8192
#define NCOLS 1024
#define P1_THREADS 256
#define P1_BLOCKS_MAX 256

// ---------------------------------------------------------------------------
// Pass 1: streaming column-sum partials of (source - target).
// Each thread owns 4 consecutive columns -> one float4 (b128) load per row per
// array; a block covers a full 4KB row per iteration (perfectly coalesced).
// Block b writes its partial colsum row to ws[b*1024 .. b*1024+1023].
// ---------------------------------------------------------------------------
__global__ __launch_bounds__(P1_THREADS) void mmd_colsum_partial(
    const float* __restrict__ src, const float* __restrict__ tgt,
    float* __restrict__ partial) {
  const int tid = threadIdx.x;        // 0..255
  const int c0  = tid * 4;            // column quartet
  const int nb  = gridDim.x;

  float ax = 0.f, ay = 0.f, az = 0.f, aw = 0.f;
  for (int r = blockIdx.x; r < NROWS; r += nb) {
    const size_t off = (size_t)r * NCOLS + c0;
    // CDNA5 speculative prefetch of the next row this block will touch.
    if (r + nb < NROWS) {
      __builtin_prefetch(src + off + (size_t)nb * NCOLS, 0, 0);
      __builtin_prefetch(tgt + off + (size_t)nb * NCOLS, 0, 0);
    }
    const float4 s = *(const float4*)(src + off);
    const float4 t = *(const float4*)(tgt + off);
    ax += s.x - t.x;
    ay += s.y - t.y;
    az += s.z - t.z;
    aw += s.w - t.w;
  }
  float4 acc = make_float4(ax, ay, az, aw);
  *(float4*)(partial + (size_t)blockIdx.x * NCOLS + c0) = acc;
}

// ---------------------------------------------------------------------------
// Pass 2: single workgroup, 512 threads = 16 wave32.
// Reduce partial[nRows][1024] over rows with V_WMMA_F32_16X16X4_F32 using an
// all-ones A(16x4): D[m][n] = sum_k B[k][n]  (column sums, exact f32).
// Wave w owns columns [64w, 64w+64) as 4 groups of 16 (B is 4x16 per step).
// Because A is all-ones, the K->lane/VGPR placement of B is irrelevant; only
// the lane->N column mapping matters (N = lane & 15, matching C/D layout).
// Each column value appears in both lane halves of C -> factor 0.5.
// ---------------------------------------------------------------------------
__global__ __launch_bounds__(512) void mmd_finalize(
    const float* __restrict__ partial, float* __restrict__ out, int nRows) {
  __shared__ float red[512];
  const int tid  = threadIdx.x;
  const int lane = tid & 31;
  const int wave = tid >> 5;       // 0..15
  const int half = lane >> 4;      // 0 or 1
  const int n    = lane & 15;      // column within group

  const v2f ones = {1.0f, 1.0f};   // A = 16x4 all-ones
  float local = 0.f;

#pragma unroll
  for (int g = 0; g < 4; ++g) {
    const int col = wave * 64 + g * 16 + n;
    v8f c = {0.f, 0.f, 0.f, 0.f, 0.f, 0.f, 0.f, 0.f};
    for (int k0 = 0; k0 < nRows; k0 += 4) {
      const int row0 = k0 + 2 * half;        // lanes 0-15: K0,K1; 16-31: K2,K3
      v2f b;
      b.x = partial[(size_t)row0 * NCOLS + col];
      b.y = partial[(size_t)(row0 + 1) * NCOLS + col];
      // 8 args: (neg_a, A, neg_b, B, c_mod, C, reuse_a, reuse_b)
      c = __builtin_amdgcn_wmma_f32_16x16x4_f32(false, ones, false, b,
                                                (short)0, c, false, false);
    }
    const float s = c[0];          // every C row holds the column sum
    local += s * s;
  }

  red[tid] = local;
  __syncthreads();
  for (int stride = 256; stride > 0; stride >>= 1) {
    if (tid < stride) red[tid] += red[tid + stride];
    __syncthreads();
  }
  if (tid == 0) {
    // sum counted each column twice; mean over N*N = 8192^2 entries
    out[0] = red[0] * (0.5f / 67108864.0f);
  }
}

extern "C" void kernel_launch(void* const* d_in, const int* in_sizes, int n_in,
                              void* d_out, int out_size, void* d_ws, size_t ws_size,
                              hipStream_t stream) {
  (void)in_sizes; (void)n_in; (void)out_size;
  const float* src = (const float*)d_in[0];
  const float* tgt = (const float*)d_in[1];
  float* out       = (float*)d_out;
  float* partial   = (float*)d_ws;

  // One partial row = 1024 floats = 4 KiB. Fit block count to ws; keep it a
  // multiple of 4 so pass-2 K-chunks of 4 divide evenly.
  int blocks = P1_BLOCKS_MAX;
  const size_t rowBytes = (size_t)NCOLS * sizeof(float);
  if ((size_t)blocks * rowBytes > ws_size) {
    blocks = (int)(ws_size / rowBytes) & ~3;
    if (blocks < 4) blocks = 4;
  }

  mmd_colsum_partial<<<blocks, P1_THREADS, 0, stream>>>(src, tgt, partial);
  mmd_finalize<<<1, 512, 0, stream>>>(partial, out, blocks);
}